// PointTransformerLayer_85461259256088
// MI455X (gfx1250) — compile-verified
//
#include <hip/hip_runtime.h>
#include <hip/hip_bf16.h>

typedef __attribute__((ext_vector_type(2))) float v2f;
typedef __attribute__((ext_vector_type(8))) float v8f;

#define DF 128
#define KNN 16
#define EPSBN 1e-5f

__device__ __forceinline__ float geluf(float x) {
    // exact erf-based GELU (torch F.gelu default)
    return 0.5f * x * (1.0f + erff(x * 0.7071067811865476f));
}

// ---------------------------------------------------------------------------
// KNN: one thread per query point, candidate points tiled through LDS,
// 16-deep sorted insertion list kept in registers.
// grid = (N/256, B), block = 256
// ---------------------------------------------------------------------------
__global__ void knn_kernel(const float* __restrict__ points, int N,
                           int* __restrict__ idxout) {
    __shared__ float tile[256 * 3];
    const int b = blockIdx.y;
    const int n = blockIdx.x * 256 + threadIdx.x;
    const float* P = points + (long)b * N * 3;

    const float qx = P[n * 3 + 0];
    const float qy = P[n * 3 + 1];
    const float qz = P[n * 3 + 2];

    float dist[KNN];
    int   ind[KNN];
#pragma unroll
    for (int j = 0; j < KNN; ++j) { dist[j] = 3.4e38f; ind[j] = 0; }

    for (int m0 = 0; m0 < N; m0 += 256) {
        __syncthreads();
        for (int i = threadIdx.x; i < 256 * 3; i += 256)
            tile[i] = P[m0 * 3 + i];
        __syncthreads();
        for (int j = 0; j < 256; ++j) {
            const float dx = qx - tile[j * 3 + 0];
            const float dy = qy - tile[j * 3 + 1];
            const float dz = qz - tile[j * 3 + 2];
            const float d2 = dx * dx + dy * dy + dz * dz;
            if (d2 < dist[KNN - 1]) {
                dist[KNN - 1] = d2; ind[KNN - 1] = m0 + j;
#pragma unroll
                for (int r = KNN - 1; r > 0; --r) {
                    if (dist[r] < dist[r - 1]) {
                        float td = dist[r]; dist[r] = dist[r - 1]; dist[r - 1] = td;
                        int   ti = ind[r];  ind[r]  = ind[r - 1];  ind[r - 1]  = ti;
                    }
                }
            }
        }
    }
    int* o = idxout + ((long)b * N + n) * KNN;
#pragma unroll
    for (int j = 0; j < KNN; ++j) o[j] = ind[j];
}

// ---------------------------------------------------------------------------
// GEMM: Y[M,128] = op(X)[M,128] @ W[128,128]^T + bias
// op(x) = identity, or gelu(x*s[k] + t[k]) (fused BatchNorm+GELU pre-op).
// In-place safe (Y may alias X): each block writes only rows it read, after
// its K-loop completes. fp32 WMMA 16x16x4 — exact fp32 accumulate.
// block = 128 (4 waves, one 16-row M-tile each), grid = M/64.
// ---------------------------------------------------------------------------
template <bool FUSE>
__global__ void gemm128_kernel(const float* __restrict__ X,
                               const float* __restrict__ W,
                               const float* __restrict__ bias,
                               const float* __restrict__ s,
                               const float* __restrict__ t,
                               float* __restrict__ Y) {
    __shared__ float ldsW[DF * DF];   // 64 KB: whole weight matrix, row-major
    const int tid = threadIdx.x;
    for (int i = tid; i < DF * DF; i += 128) ldsW[i] = W[i];
    __syncthreads();

    const int wave = tid >> 5;
    const int lane = tid & 31;
    const int hi = lane >> 4;     // 0: K pair {0,1}, 1: K pair {2,3}
    const int lo = lane & 15;

    const long Mb = ((long)blockIdx.x * 4 + wave) * 16;
    const float* Xrow = X + (Mb + lo) * DF;

    v8f acc[8] = {};
    for (int kb = 0; kb < DF; kb += 4) {
        const int kk = kb + 2 * hi;
        v2f a = *(const v2f*)(Xrow + kk);
        if (FUSE) {
            a.x = geluf(a.x * s[kk] + t[kk]);
            a.y = geluf(a.y * s[kk + 1] + t[kk + 1]);
        }
#pragma unroll
        for (int nt = 0; nt < 8; ++nt) {
            const v2f bfrag = *(const v2f*)(&ldsW[(nt * 16 + lo) * DF + kk]);
            acc[nt] = __builtin_amdgcn_wmma_f32_16x16x4_f32(
                false, a, false, bfrag, (short)0, acc[nt], false, false);
        }
    }
#pragma unroll
    for (int nt = 0; nt < 8; ++nt) {
        const int col = nt * 16 + lo;
        const float bb = bias[col];
#pragma unroll
        for (int r = 0; r < 8; ++r) {
            const long row = Mb + r + 8 * hi;
            Y[row * DF + col] = acc[nt][r] + bb;
        }
    }
}

// ---------------------------------------------------------------------------
// zero a float region (stats accumulators) — must run every launch.
// ---------------------------------------------------------------------------
__global__ void zero_kernel(float* p, int n) {
    int i = blockIdx.x * blockDim.x + threadIdx.x;
    if (i < n) p[i] = 0.0f;
}

// ---------------------------------------------------------------------------
// pos1 = (points[n] - points[idx]) @ Wd1^T + bd1   [M2,3], plus BN stats
// over the 3 channels (LDS float atomics -> global float atomics).
// grid = M2/256, block = 256 (one row per thread)
// ---------------------------------------------------------------------------
__global__ void posprep_kernel(const float* __restrict__ points,
                               const int* __restrict__ idx,
                               const float* __restrict__ Wd1,
                               const float* __restrict__ bd1,
                               float* __restrict__ pos1,
                               float* __restrict__ gsum3,
                               float* __restrict__ gsq3, int N) {
    __shared__ float accum[6];
    if (threadIdx.x < 6) accum[threadIdx.x] = 0.0f;
    __syncthreads();

    const long r  = (long)blockIdx.x * 256 + threadIdx.x;
    const long bn = r >> 4;
    const long b  = bn / N;
    const int  nb = idx[r];

    const float* pq = points + bn * 3;
    const float* pk = points + (b * N + nb) * 3;
    const float d0 = pq[0] - pk[0];
    const float d1 = pq[1] - pk[1];
    const float d2 = pq[2] - pk[2];

#pragma unroll
    for (int o = 0; o < 3; ++o) {
        const float y = d0 * Wd1[o * 3 + 0] + d1 * Wd1[o * 3 + 1] +
                        d2 * Wd1[o * 3 + 2] + bd1[o];
        pos1[r * 3 + o] = y;
        atomicAdd(&accum[o], y);
        atomicAdd(&accum[3 + o], y * y);
    }
    __syncthreads();
    if (threadIdx.x < 3) {
        atomicAdd(&gsum3[threadIdx.x], accum[threadIdx.x]);
        atomicAdd(&gsq3[threadIdx.x], accum[3 + threadIdx.x]);
    }
}

// ---------------------------------------------------------------------------
// per-channel sum / sum-of-squares over X[M,128]
// block = 256 (two row-groups x 128 channels), grid-stride over rows
// ---------------------------------------------------------------------------
__global__ void stats128_kernel(const float* __restrict__ X, long Mrows,
                                float* __restrict__ gsum,
                                float* __restrict__ gsq) {
    __shared__ float ssum[DF];
    __shared__ float ssq[DF];
    const int t = threadIdx.x;
    const int c = t & (DF - 1);
    const int half = t >> 7;

    float ls = 0.0f, lq = 0.0f;
    for (long r = (long)blockIdx.x * 2 + half; r < Mrows;
         r += (long)gridDim.x * 2) {
        const float x = X[r * DF + c];
        ls += x; lq += x * x;
    }
    if (half == 0) { ssum[c] = ls; ssq[c] = lq; }
    __syncthreads();
    if (half == 1) { ssum[c] += ls; ssq[c] += lq; }
    __syncthreads();
    if (half == 0) {
        atomicAdd(&gsum[c], ssum[c]);
        atomicAdd(&gsq[c], ssq[c]);
    }
}

// ---------------------------------------------------------------------------
// fold BN stats into scale/shift: y = x*s + t
// ---------------------------------------------------------------------------
__global__ void finalize_kernel(const float* __restrict__ gsum,
                                const float* __restrict__ gsq, float cnt,
                                const float* __restrict__ gamma,
                                const float* __restrict__ beta,
                                float* __restrict__ s, float* __restrict__ t,
                                int C) {
    const int c = threadIdx.x;
    if (c < C) {
        const float m = gsum[c] / cnt;
        const float v = gsq[c] / cnt - m * m;
        const float r = rsqrtf(v + EPSBN);
        const float sc = r * gamma[c];
        s[c] = sc;
        t[c] = beta[c] - m * sc;
    }
}

// ---------------------------------------------------------------------------
// gamma0 = Q - gather(K) + pos2, where pos2 = gelu(bn(pos1)) @ Wd2^T + bd2.
// Stores pos2 (reused at the end) and gamma0. grid = M2, block = 128.
// ---------------------------------------------------------------------------
__global__ void gamma0_kernel(const float* __restrict__ pos1,
                              const float* __restrict__ sd,
                              const float* __restrict__ td,
                              const float* __restrict__ Wd2,
                              const float* __restrict__ bd2,
                              const float* __restrict__ Q,
                              const float* __restrict__ Kf,
                              const int* __restrict__ idx,
                              float* __restrict__ pos2,
                              float* __restrict__ gamma0, int N) {
    __shared__ float h[3];
    const long r = blockIdx.x;
    const int  d = threadIdx.x;
    if (d < 3) {
        const float x = pos1[r * 3 + d];
        h[d] = geluf(x * sd[d] + td[d]);
    }
    __syncthreads();
    const float p2 = h[0] * Wd2[d * 3 + 0] + h[1] * Wd2[d * 3 + 1] +
                     h[2] * Wd2[d * 3 + 2] + bd2[d];
    const long bn = r >> 4;
    const long b  = bn / N;
    const int  nb = idx[r];
    const float g0 = Q[bn * DF + d] - Kf[(b * N + nb) * DF + d] + p2;
    pos2[r * DF + d]   = p2;
    gamma0[r * DF + d] = g0;
}

// ---------------------------------------------------------------------------
// softmax over k neighbors per (b,n,d), then out = sum_k rho * (V_gath+pos2)
// grid = B*N, block = 128
// ---------------------------------------------------------------------------
__global__ void softmax_out_kernel(const float* __restrict__ gamma2,
                                   const float* __restrict__ pos2,
                                   const float* __restrict__ Vf,
                                   const int* __restrict__ idx,
                                   float* __restrict__ out, int N) {
    const long bn = blockIdx.x;
    const int  d  = threadIdx.x;
    const long b  = bn / N;

    float g[KNN];
    float mx = -3.4e38f;
#pragma unroll
    for (int k = 0; k < KNN; ++k) {
        g[k] = gamma2[(bn * KNN + k) * DF + d];
        mx = fmaxf(mx, g[k]);
    }
    float sum = 0.0f;
#pragma unroll
    for (int k = 0; k < KNN; ++k) { g[k] = expf(g[k] - mx); sum += g[k]; }
    const float inv = 1.0f / sum;
    float acc = 0.0f;
#pragma unroll
    for (int k = 0; k < KNN; ++k) {
        const int nb = idx[bn * KNN + k];
        const float val = Vf[(b * N + nb) * DF + d] +
                          pos2[(bn * KNN + k) * DF + d];
        acc += g[k] * inv * val;
    }
    out[bn * DF + d] = acc;
}

// ---------------------------------------------------------------------------
extern "C" void kernel_launch(void* const* d_in, const int* in_sizes, int n_in,
                              void* d_out, int out_size, void* d_ws, size_t ws_size,
                              hipStream_t stream) {
    const float* feats  = (const float*)d_in[0];
    const float* points = (const float*)d_in[1];
    const float* Wq = (const float*)d_in[2];  const float* bq = (const float*)d_in[3];
    const float* Wk = (const float*)d_in[4];  const float* bk = (const float*)d_in[5];
    const float* Wv = (const float*)d_in[6];  const float* bv = (const float*)d_in[7];
    const float* Wd1 = (const float*)d_in[8]; const float* bd1 = (const float*)d_in[9];
    const float* Wd2 = (const float*)d_in[10]; const float* bd2 = (const float*)d_in[11];
    const float* Wg1 = (const float*)d_in[12]; const float* bg1 = (const float*)d_in[13];
    const float* Wg2 = (const float*)d_in[14]; const float* bg2 = (const float*)d_in[15];
    const float* gd  = (const float*)d_in[16]; const float* betad  = (const float*)d_in[17];
    const float* gg1 = (const float*)d_in[18]; const float* betag1 = (const float*)d_in[19];
    const float* gg2 = (const float*)d_in[20]; const float* betag2 = (const float*)d_in[21];

    const int  N  = 8192;
    const long BN = (long)in_sizes[1] / 3;      // B*N
    const int  B  = (int)(BN / N);
    const long M2 = BN * KNN;                   // B*N*k rows

    // workspace layout (all chunks 16B-aligned)
    char* w = (char*)d_ws;
    int*   idxb = (int*)w;      w += BN * KNN * sizeof(int);
    float* Qb   = (float*)w;    w += BN * DF * sizeof(float);
    float* Kb   = (float*)w;    w += BN * DF * sizeof(float);
    float* Vb   = (float*)w;    w += BN * DF * sizeof(float);
    float* pos1 = (float*)w;    w += M2 * 3 * sizeof(float);
    float* pos2 = (float*)w;    w += M2 * DF * sizeof(float);
    float* G    = (float*)w;    w += M2 * DF * sizeof(float);  // gamma0/1/2 in place
    float* gsum3 = (float*)w;   w += 4 * sizeof(float);
    float* gsq3  = (float*)w;   w += 4 * sizeof(float);
    float* s3    = (float*)w;   w += 4 * sizeof(float);
    float* t3    = (float*)w;   w += 4 * sizeof(float);
    float* gsum128 = (float*)w; w += DF * sizeof(float);
    float* gsq128  = (float*)w; w += DF * sizeof(float);
    float* s128    = (float*)w; w += DF * sizeof(float);
    float* t128    = (float*)w; w += DF * sizeof(float);

    const float cnt = (float)M2;  // BN stats count (B*N*k) for both BN sites

    // 1) KNN indices
    knn_kernel<<<dim3(N / 256, B), 256, 0, stream>>>(points, N, idxb);

    // 2) Q, K, V projections (lin before gather — equivalent, 16x cheaper)
    gemm128_kernel<false><<<(int)(BN / 64), 128, 0, stream>>>(feats, Wq, bq, nullptr, nullptr, Qb);
    gemm128_kernel<false><<<(int)(BN / 64), 128, 0, stream>>>(feats, Wk, bk, nullptr, nullptr, Kb);
    gemm128_kernel<false><<<(int)(BN / 64), 128, 0, stream>>>(feats, Wv, bv, nullptr, nullptr, Vb);

    // 3) pos MLP stage 1 + BN(3) stats
    zero_kernel<<<1, 64, 0, stream>>>(gsum3, 8);        // gsum3+gsq3
    posprep_kernel<<<(int)(M2 / 256), 256, 0, stream>>>(points, idxb, Wd1, bd1,
                                                        pos1, gsum3, gsq3, N);
    finalize_kernel<<<1, 128, 0, stream>>>(gsum3, gsq3, cnt, gd, betad, s3, t3, 3);

    // 4) pos2 + gamma0 = Q - gather(K) + pos2
    gamma0_kernel<<<(int)M2, 128, 0, stream>>>(pos1, s3, t3, Wd2, bd2,
                                               Qb, Kb, idxb, pos2, G, N);

    // 5) BN1 stats -> fused BN+GELU+GEMM (gamma0 -> gamma1, in place)
    zero_kernel<<<1, 256, 0, stream>>>(gsum128, 2 * DF);
    stats128_kernel<<<2048, 256, 0, stream>>>(G, M2, gsum128, gsq128);
    finalize_kernel<<<1, 128, 0, stream>>>(gsum128, gsq128, cnt, gg1, betag1, s128, t128, DF);
    gemm128_kernel<true><<<(int)(M2 / 64), 128, 0, stream>>>(G, Wg1, bg1, s128, t128, G);

    // 6) BN2 stats -> fused BN+GELU+GEMM (gamma1 -> gamma2, in place)
    zero_kernel<<<1, 256, 0, stream>>>(gsum128, 2 * DF);
    stats128_kernel<<<2048, 256, 0, stream>>>(G, M2, gsum128, gsq128);
    finalize_kernel<<<1, 128, 0, stream>>>(gsum128, gsq128, cnt, gg2, betag2, s128, t128, DF);
    gemm128_kernel<true><<<(int)(M2 / 64), 128, 0, stream>>>(G, Wg2, bg2, s128, t128, G);

    // 7) softmax over k + weighted sum of (gather(V) + pos2)
    softmax_out_kernel<<<(int)BN, 128, 0, stream>>>(G, pos2, Vb, idxb,
                                                    (float*)d_out, N);
}